// MixConConbaMixer_90074054131939
// MI455X (gfx1250) — compile-verified
//
#include <hip/hip_runtime.h>
#include <hip/hip_bf16.h>
#include <math.h>

// Problem constants (from reference setup_inputs)
#define BSZ   8
#define SEQ   2048
#define HD    512
#define ID    512
#define NS    512
#define RD    32
#define KC    4
#define MROWS (BSZ * SEQ)     // 16384 token rows
#define PDIM  (RD + 2 * NS)   // 1056
#define TWOI  (2 * ID)        // 1024

// GEMM tiling: block = 256 threads = 8 waves; block tile 128x32; K-step 16.
#define GEMM_BM 128
#define GEMM_BN 32
#define GEMM_KT 16
#define WPAD    (GEMM_KT + 4)  // 20 floats (80B) rows: 16B-aligned, bank-conflict-free

typedef __attribute__((ext_vector_type(2))) float v2f;
typedef __attribute__((ext_vector_type(4))) float v4f;
typedef __attribute__((ext_vector_type(8))) float v8f;
typedef __attribute__((ext_vector_type(4))) int   v4i;

#if __has_builtin(__builtin_amdgcn_global_load_async_to_lds_b128) && \
    __has_builtin(__builtin_amdgcn_s_wait_asynccnt)
#define USE_ASYNC_LDS 1
#else
#define USE_ASYNC_LDS 0
#endif

__device__ __forceinline__ float silu_f(float x) { return x / (1.f + __expf(-x)); }
__device__ __forceinline__ float softplus_f(float x) {
  return (x > 20.f) ? x : log1pf(__expf(x));
}

// ---------------------------------------------------------------------------
// C[M,N] = A[M,K] @ W[N,K]^T  (fp32 row-major) via V_WMMA_F32_16X16X4_F32.
// Block computes a 128x32 tile: 8 waves, each wave a 16x32 strip (2 accumulators,
// A-frag reused for 2 WMMAs). The shared 32x16 W-tile is staged in LDS (double-
// buffered) via GLOBAL_LOAD_ASYNC_TO_LDS_B128 + s_wait_asynccnt when available.
// ---------------------------------------------------------------------------
__global__ __launch_bounds__(256) void gemm_f32_wmma(const float* __restrict__ A,
                                                     const float* __restrict__ W,
                                                     float* __restrict__ C,
                                                     int M, int N, int K) {
  __shared__ float wbuf[2][GEMM_BN][WPAD];

  const int tid    = threadIdx.x;
  const int waveId = tid >> 5;          // 0..7
  const int lane   = tid & 31;
  const int half   = lane >> 4;
  const int r      = lane & 15;

  const int nBN = N / GEMM_BN;
  const int bm  = blockIdx.x / nBN;
  const int bn  = blockIdx.x % nBN;

  // A fragment source: this lane's row, k offset 2*half (+j) per sub-step.
  const float* __restrict__ Arow =
      A + (size_t)(bm * GEMM_BM + waveId * 16 + r) * K + 2 * half;

  // W staging: thread t (t<128) copies 16B: W[bn*32 + t/4][k0 + (t%4)*4 .. +3]
  const int  wrow    = tid >> 2;        // 0..63, only <32 active
  const int  wk4     = (tid & 3) * 4;   // 0,4,8,12
  const bool doStage = (wrow < GEMM_BN);
  const float* __restrict__ Wg = W + (size_t)(bn * GEMM_BN + wrow) * K + wk4;

  v8f acc0 = {}, acc1 = {};

  auto stage = [&](int buf, int k0) {
    if (doStage) {
      const float* src = Wg + k0;
      float*       dst = &wbuf[buf][wrow][wk4];
#if USE_ASYNC_LDS
      // signature (from probe diagnostic): (v4i AS1*, v4i AS3*, imm off, imm cpol)
      __builtin_amdgcn_global_load_async_to_lds_b128(
          (__attribute__((address_space(1))) v4i*)src,
          (__attribute__((address_space(3))) v4i*)dst, 0, 0);
#else
      *(v4f*)dst = *(const v4f*)src;
#endif
    }
  };
  auto waitStage = [&]() {
#if USE_ASYNC_LDS
    __builtin_amdgcn_s_wait_asynccnt(0);
#endif
    __syncthreads();
  };

  stage(0, 0);
  waitStage();

  int cur = 0;
  for (int k0 = 0; k0 < K; k0 += GEMM_KT) {
    if (k0 + GEMM_KT < K) stage(cur ^ 1, k0 + GEMM_KT);  // prefetch next W tile

    #pragma unroll
    for (int kk = 0; kk < GEMM_KT; kk += 4) {
      v2f a, b0, b1;
      a[0] = Arow[k0 + kk];
      a[1] = Arow[k0 + kk + 1];
      const int kl = kk + 2 * half;
      b0[0] = wbuf[cur][r][kl];       b0[1] = wbuf[cur][r][kl + 1];
      b1[0] = wbuf[cur][16 + r][kl];  b1[1] = wbuf[cur][16 + r][kl + 1];
      acc0 = __builtin_amdgcn_wmma_f32_16x16x4_f32(false, a, false, b0, (short)0,
                                                   acc0, false, false);
      acc1 = __builtin_amdgcn_wmma_f32_16x16x4_f32(false, a, false, b1, (short)0,
                                                   acc1, false, false);
    }
    waitStage();  // next buffer ready + everyone done reading cur
    cur ^= 1;
  }

  float* __restrict__ C0 =
      C + (size_t)(bm * GEMM_BM + waveId * 16 + 8 * half) * N + bn * GEMM_BN + r;
  #pragma unroll
  for (int v = 0; v < 8; ++v) {
    C0[(size_t)v * N]      = acc0[v];
    C0[(size_t)v * N + 16] = acc1[v];
  }
}

// ---------------------------------------------------------------------------
// Causal depthwise conv1d (K=4) + SiLU.  x half of proj is columns [0, I).
// ---------------------------------------------------------------------------
__global__ void conv_silu_kernel(const float* __restrict__ proj,
                                 const float* __restrict__ cw,
                                 const float* __restrict__ cb,
                                 float* __restrict__ xconv) {
  int id = blockIdx.x * blockDim.x + threadIdx.x;
  if (id >= MROWS * ID) return;
  int i = id % ID;
  int m = id / ID;
  int b = m / SEQ, s = m % SEQ;
  float acc = cb[i];
  #pragma unroll
  for (int k = 0; k < KC; ++k) {
    int sp = s - (KC - 1) + k;
    if (sp >= 0)
      acc = fmaf(cw[i * KC + k], proj[(size_t)(b * SEQ + sp) * TWOI + i], acc);
  }
  xconv[id] = silu_f(acc);
}

// ---------------------------------------------------------------------------
// Fused per-row RMSNorm of the 3 parts of ssm_p: dt[0,32), B[32,544), C[544,1056).
// One wave32 per row; butterfly shuffle reductions (wave32-correct).
// ---------------------------------------------------------------------------
__global__ void rmsnorm3_kernel(float* __restrict__ ssm, const float* __restrict__ dtw,
                                const float* __restrict__ bw, const float* __restrict__ cw) {
  int w = blockIdx.x * (blockDim.x >> 5) + (threadIdx.x >> 5);
  if (w >= MROWS) return;
  int lane = threadIdx.x & 31;
  float* row = ssm + (size_t)w * PDIM;

  // dt part (32 elements, one per lane)
  float xv = row[lane];
  float ss = xv * xv;
  #pragma unroll
  for (int o = 16; o; o >>= 1) ss += __shfl_xor(ss, o, 32);
  row[lane] = dtw[lane] * xv * rsqrtf(ss * (1.f / RD) + 1e-6f);

  // B part (512 elements, 16 per lane)
  float vals[16];
  ss = 0.f;
  #pragma unroll
  for (int j = 0; j < 16; ++j) {
    vals[j] = row[RD + lane + 32 * j];
    ss = fmaf(vals[j], vals[j], ss);
  }
  #pragma unroll
  for (int o = 16; o; o >>= 1) ss += __shfl_xor(ss, o, 32);
  float sc = rsqrtf(ss * (1.f / NS) + 1e-6f);
  #pragma unroll
  for (int j = 0; j < 16; ++j)
    row[RD + lane + 32 * j] = bw[lane + 32 * j] * vals[j] * sc;

  // C part (512 elements, 16 per lane)
  ss = 0.f;
  #pragma unroll
  for (int j = 0; j < 16; ++j) {
    vals[j] = row[RD + NS + lane + 32 * j];
    ss = fmaf(vals[j], vals[j], ss);
  }
  #pragma unroll
  for (int o = 16; o; o >>= 1) ss += __shfl_xor(ss, o, 32);
  sc = rsqrtf(ss * (1.f / NS) + 1e-6f);
  #pragma unroll
  for (int j = 0; j < 16; ++j)
    row[RD + NS + lane + 32 * j] = cw[lane + 32 * j] * vals[j] * sc;
}

// ---------------------------------------------------------------------------
// dt_full[m,i] = softplus(dt_norm[m,:] . dt_proj_w[i,:] + dt_proj_b[i])  (K=32)
// ---------------------------------------------------------------------------
__global__ void dtproj_kernel(const float* __restrict__ ssm, const float* __restrict__ dw,
                              const float* __restrict__ db, float* __restrict__ dtf) {
  int id = blockIdx.x * blockDim.x + threadIdx.x;
  if (id >= MROWS * ID) return;
  int i = id % ID;
  int m = id / ID;
  const float* row = ssm + (size_t)m * PDIM;  // dt part is at offset 0
  float a = db[i];
  #pragma unroll
  for (int r2 = 0; r2 < RD; ++r2) a = fmaf(row[r2], dw[i * RD + r2], a);
  dtf[id] = softplus_f(a);
}

// ---------------------------------------------------------------------------
// Selective scan: one wave32 per (b,i); each lane owns 16 of the N=512 states
// in registers. Per step: dA = exp(dt*A), state = state*dA + dt*x*B, y = state.C
// reduced across the wave. All I waves of a batch reuse the same B/C rows -> L2.
// global_prefetch of step s+1's B/C rows hides latency behind the exp/fma chain.
// ---------------------------------------------------------------------------
__global__ void scan_kernel(const float* __restrict__ ssm, const float* __restrict__ dtf,
                            const float* __restrict__ xconv,
                            const float* __restrict__ A_log, float* __restrict__ ysc) {
  int w = blockIdx.x * (blockDim.x >> 5) + (threadIdx.x >> 5);
  if (w >= BSZ * ID) return;
  int lane = threadIdx.x & 31;
  int b = w / ID, i = w % ID;

  float a_[16], st[16];
  #pragma unroll
  for (int j = 0; j < 16; ++j) {
    a_[j] = -__expf(A_log[(size_t)i * NS + lane + 32 * j]);
    st[j] = 0.f;
  }

  for (int s = 0; s < SEQ; ++s) {
    size_t m = (size_t)b * SEQ + s;
    float dt  = dtf[m * ID + i];     // wave-uniform scalar
    float xv  = xconv[m * ID + i];   // wave-uniform scalar
    float dtx = dt * xv;
    const float* __restrict__ brow = ssm + m * PDIM + RD;
    const float* __restrict__ crow = brow + NS;
    if (s + 1 < SEQ) {  // prefetch next step's B/C rows (2KB each, 64B per lane)
      __builtin_prefetch(brow + PDIM + lane * 16, 0, 1);
      __builtin_prefetch(crow + PDIM + lane * 16, 0, 1);
    }
    float acc = 0.f;
    #pragma unroll
    for (int j = 0; j < 16; ++j) {
      float dA = __expf(dt * a_[j]);
      float bv = brow[lane + 32 * j];   // coalesced 128B per j
      float cv = crow[lane + 32 * j];
      st[j] = fmaf(st[j], dA, dtx * bv);
      acc   = fmaf(st[j], cv, acc);
    }
    #pragma unroll
    for (int o = 16; o; o >>= 1) acc += __shfl_xor(acc, o, 32);
    if (lane == 0) ysc[m * ID + i] = acc;
  }
}

// ---------------------------------------------------------------------------
// y = (y_scan + D*x_conv) * silu(gate);  gate is columns [I, 2I) of proj.
// In-place on ysc.
// ---------------------------------------------------------------------------
__global__ void gate_kernel(float* __restrict__ ysc, const float* __restrict__ xconv,
                            const float* __restrict__ proj, const float* __restrict__ Dv) {
  int id = blockIdx.x * blockDim.x + threadIdx.x;
  if (id >= MROWS * ID) return;
  int i = id % ID;
  int m = id / ID;
  float yv = ysc[id] + Dv[i] * xconv[id];
  float g  = proj[(size_t)m * TWOI + ID + i];
  ysc[id] = yv * silu_f(g);
}

// ---------------------------------------------------------------------------
// Conba epilogues (in-place where legal)
// ---------------------------------------------------------------------------
__global__ void ew_scale_kernel(float* __restrict__ o, const float* __restrict__ t,
                                const float* __restrict__ bias) {
  int id = blockIdx.x * blockDim.x + threadIdx.x;
  if (id >= MROWS * HD) return;
  int h = id % HD;
  o[id] = o[id] * (t[id] + bias[h]);
}
__global__ void ew_delay_kernel(float* __restrict__ o, const float* __restrict__ t,
                                const float* __restrict__ bias) {
  int id = blockIdx.x * blockDim.x + threadIdx.x;
  if (id >= MROWS * HD) return;
  int h = id % HD;
  o[id] = o[id] + t[id] + bias[h];
}
__global__ void ew_ctrl_kernel(const float* __restrict__ o, const float* __restrict__ t,
                               float* __restrict__ dout) {
  int id = blockIdx.x * blockDim.x + threadIdx.x;
  if (id >= MROWS * HD) return;
  dout[id] = o[id] - t[id];
}

// ---------------------------------------------------------------------------
extern "C" void kernel_launch(void* const* d_in, const int* in_sizes, int n_in,
                              void* d_out, int out_size, void* d_ws, size_t ws_size,
                              hipStream_t stream) {
  (void)in_sizes; (void)n_in; (void)out_size; (void)ws_size;

  const float* hidden    = (const float*)d_in[0];   // [B,S,H]
  const float* in_proj_w = (const float*)d_in[1];   // [2I,H]
  const float* conv_w    = (const float*)d_in[2];   // [I,K]
  const float* conv_b    = (const float*)d_in[3];   // [I]
  const float* x_proj_w  = (const float*)d_in[4];   // [R+2N, I]
  const float* dt_proj_w = (const float*)d_in[5];   // [I,R]
  const float* dt_proj_b = (const float*)d_in[6];   // [I]
  const float* dt_ln_w   = (const float*)d_in[7];   // [R]
  const float* b_ln_w    = (const float*)d_in[8];   // [N]
  const float* c_ln_w    = (const float*)d_in[9];   // [N]
  const float* A_log     = (const float*)d_in[10];  // [I,N]
  const float* Dvec      = (const float*)d_in[11];  // [I]
  const float* out_proj_w= (const float*)d_in[12];  // [H,I]
  const float* scale_w   = (const float*)d_in[13];  // [H,H]
  const float* scale_b   = (const float*)d_in[14];  // [H]
  const float* delay_w   = (const float*)d_in[15];  // [H,H]
  const float* delay_b   = (const float*)d_in[16];  // [H]
  const float* ctrl_K    = (const float*)d_in[17];  // [I,I]
  float* out_final = (float*)d_out;

  // Workspace layout (floats): ~304 MB total
  float* ws    = (float*)d_ws;
  float* proj  = ws;                                   // [M, 2I]   16.78M
  float* xconv = proj  + (size_t)MROWS * TWOI;         // [M, I]     8.39M
  float* ssm   = xconv + (size_t)MROWS * ID;           // [M, P]    17.30M
  float* dtf   = ssm   + (size_t)MROWS * PDIM;         // [M, I]     8.39M
  float* ysc   = dtf   + (size_t)MROWS * ID;           // [M, I]     8.39M
  float* out0  = ysc   + (size_t)MROWS * ID;           // [M, H]     8.39M
  float* tbuf  = out0  + (size_t)MROWS * HD;           // [M, H]     8.39M

  const int TPB = 256;
  const int ewBlocksI  = (MROWS * ID + TPB - 1) / TPB;
  const int ewBlocksH  = (MROWS * HD + TPB - 1) / TPB;
  const int rowBlocks  = (MROWS + 7) / 8;              // 8 waves per block
  const int scanBlocks = (BSZ * ID + 7) / 8;

  auto gemm = [&](const float* A, const float* W, float* C, int M, int N, int K) {
    int blocks = (M / GEMM_BM) * (N / GEMM_BN);        // all shapes divide evenly
    gemm_f32_wmma<<<blocks, TPB, 0, stream>>>(A, W, C, M, N, K);
  };

  // 1. in_proj: proj[M,2I] = hidden[M,H] @ in_proj_w[2I,H]^T
  gemm(hidden, in_proj_w, proj, MROWS, TWOI, HD);
  // 2. causal depthwise conv + silu
  conv_silu_kernel<<<ewBlocksI, TPB, 0, stream>>>(proj, conv_w, conv_b, xconv);
  // 3. x_proj: ssm[M,P] = xconv[M,I] @ x_proj_w[P,I]^T
  gemm(xconv, x_proj_w, ssm, MROWS, PDIM, ID);
  // 4. three-part RMSNorm in place
  rmsnorm3_kernel<<<rowBlocks, TPB, 0, stream>>>(ssm, dt_ln_w, b_ln_w, c_ln_w);
  // 5. dt projection + softplus
  dtproj_kernel<<<ewBlocksI, TPB, 0, stream>>>(ssm, dt_proj_w, dt_proj_b, dtf);
  // 6. selective scan
  scan_kernel<<<scanBlocks, TPB, 0, stream>>>(ssm, dtf, xconv, A_log, ysc);
  // 7. skip + gate (in place on ysc)
  gate_kernel<<<ewBlocksI, TPB, 0, stream>>>(ysc, xconv, proj, Dvec);
  // 8. out_proj: out0[M,H] = ysc[M,I] @ out_proj_w[H,I]^T
  gemm(ysc, out_proj_w, out0, MROWS, HD, ID);
  // 9-10. out *= (out @ scale_w^T + scale_b)
  gemm(out0, scale_w, tbuf, MROWS, HD, HD);
  ew_scale_kernel<<<ewBlocksH, TPB, 0, stream>>>(out0, tbuf, scale_b);
  // 11-12. out += (out @ delay_w^T + delay_b)
  gemm(out0, delay_w, tbuf, MROWS, HD, HD);
  ew_delay_kernel<<<ewBlocksH, TPB, 0, stream>>>(out0, tbuf, delay_b);
  // 13-14. out -= out @ ctrl_K^T
  gemm(out0, ctrl_K, tbuf, MROWS, HD, HD);
  ew_ctrl_kernel<<<ewBlocksH, TPB, 0, stream>>>(out0, tbuf, out_final);
}